// OriginalGINConv_29643864277575
// MI455X (gfx1250) — compile-verified
//
#include <hip/hip_runtime.h>
#include <hip/hip_bf16.h>

typedef __attribute__((ext_vector_type(2))) float v2f;
typedef __attribute__((ext_vector_type(8))) float v8f;

#define N_NODES   100000
#define N_EDGES   1600000
#define D_IN      64
#define D_EMB     128
#define BN_EPS    1e-5f

// ---------------------------------------------------------------------------
// K0: zero agg + BN stat accumulators (ws is not re-poisoned between replays)
// ---------------------------------------------------------------------------
__global__ void zero_ws_kernel(float* __restrict__ p, long long n) {
    long long i = (long long)blockIdx.x * blockDim.x + threadIdx.x;
    long long stride = (long long)gridDim.x * blockDim.x;
    for (; i < n; i += stride) p[i] = 0.0f;
}

// ---------------------------------------------------------------------------
// K1: agg[dst] += x[src]  — float4 gathers, 4 atomics/thread; agg/x L2-resident
// ---------------------------------------------------------------------------
__global__ void scatter_add_kernel(const float* __restrict__ x,
                                   const int* __restrict__ edge_index,
                                   float* __restrict__ agg) {
    long long t = (long long)blockIdx.x * blockDim.x + threadIdx.x;
    if (t >= (long long)N_EDGES * (D_IN / 4)) return;
    int e  = (int)(t >> 4);       // 16 float4 per edge
    int f4 = (int)(t & 15);
    int src = edge_index[e];
    int dst = edge_index[N_EDGES + e];
    const float4 v = *(const float4*)(x + (size_t)src * D_IN + f4 * 4);
    float* d = agg + (size_t)dst * D_IN + f4 * 4;
    atomicAdd(d + 0, v.x);
    atomicAdd(d + 1, v.y);
    atomicAdd(d + 2, v.z);
    atomicAdd(d + 3, v.w);
}

// ---------------------------------------------------------------------------
// K2: h1 = (x + agg) @ W1 + b1  via v_wmma_f32_16x16x4_f32, fused BN stats.
// W1 in LDS as swizzled row-pairs (v2f) -> conflict-free ds_load_2addr_b64.
// All 8 B fragments prefetched per k-step so LDS latency overlaps the WMMAs.
// ---------------------------------------------------------------------------
__global__ __launch_bounds__(256) void gemm1_bn_kernel(
        const float* __restrict__ x, const float* __restrict__ agg,
        const float* __restrict__ W1, const float* __restrict__ b1,
        float* __restrict__ h1, float* __restrict__ colsum,
        float* __restrict__ colsumsq) {
    __shared__ v2f   sP[(D_IN / 2) * D_EMB];   // 32 KB, paired rows
    __shared__ float sb[D_EMB];

    int tid = threadIdx.x;
    for (int i = tid; i < D_IN * D_EMB; i += 256) {
        int k = i >> 7, n = i & 127, p = k >> 1;
        int nsw = n ^ ((p & 1) << 4);          // bank swizzle
        ((float*)&sP[p * D_EMB + nsw])[k & 1] = W1[i];
    }
    if (tid < D_EMB) sb[tid] = b1[tid];
    __syncthreads();

    int wave = tid >> 5;
    int lane = tid & 31;
    int half = lane >> 4;                      // 0: K%4 in {0,1}; 1: {2,3}
    int l16  = lane & 15;
    int row0 = blockIdx.x * 128 + wave * 16;
    int arow = row0 + l16;
    float amask = (arow < N_NODES) ? 1.0f : 0.0f;
    int    crow = (arow < N_NODES) ? arow : (N_NODES - 1);   // clamped: safe load
    const v2f* x2 = (const v2f*)(x   + (size_t)crow * D_IN);
    const v2f* g2 = (const v2f*)(agg + (size_t)crow * D_IN);

    v8f acc[8] = {};

    #pragma unroll 2
    for (int kk = 0; kk < D_IN / 4; ++kk) {
        int p  = 2 * kk + half;
        int sw = p & 1;
        v2f a = x2[p] + g2[p];
        a.x *= amask; a.y *= amask;
        v2f bf[8];
        #pragma unroll
        for (int nt = 0; nt < 8; ++nt)
            bf[nt] = sP[p * D_EMB + ((nt ^ sw) * 16) + l16];
        #pragma unroll
        for (int nt = 0; nt < 8; ++nt)
            acc[nt] = __builtin_amdgcn_wmma_f32_16x16x4_f32(
                false, a, false, bf[nt], (short)0, acc[nt], false, false);
    }

    #pragma unroll
    for (int nt = 0; nt < 8; ++nt) {
        int col = nt * 16 + l16;
        float bias = sb[col];
        float s = 0.0f, sq = 0.0f;
        #pragma unroll
        for (int v = 0; v < 8; ++v) {
            int r = row0 + v + half * 8;       // C/D layout: VGPR v -> rows v / v+8
            float val = acc[nt][v] + bias;
            if (r < N_NODES) {
                h1[(size_t)r * D_EMB + col] = val;
                s  += val;
                sq += val * val;
            }
        }
        atomicAdd(&colsum[col], s);
        atomicAdd(&colsumsq[col], sq);
    }
}

// ---------------------------------------------------------------------------
// K3: fold BN mean/var + gamma/beta into per-column scale/shift
// ---------------------------------------------------------------------------
__global__ void bn_finalize_kernel(const float* __restrict__ colsum,
                                   const float* __restrict__ colsumsq,
                                   const float* __restrict__ gamma,
                                   const float* __restrict__ beta,
                                   float* __restrict__ scale,
                                   float* __restrict__ shift) {
    int i = threadIdx.x;   // 128 threads
    float inv_n = 1.0f / (float)N_NODES;
    float mean = colsum[i] * inv_n;
    float var  = colsumsq[i] * inv_n - mean * mean;
    float rstd = rsqrtf(var + BN_EPS);
    float sc = rstd * gamma[i];
    scale[i] = sc;
    shift[i] = beta[i] - mean * sc;
}

// ---------------------------------------------------------------------------
// K4: out = relu(h1*scale + shift) @ W2 + b2  via v_wmma_f32_16x16x4_f32.
// W2 (64 KB) in LDS as swizzled row-pairs; branchless A; fused BN+ReLU.
// ---------------------------------------------------------------------------
__global__ __launch_bounds__(256) void gemm2_kernel(
        const float* __restrict__ h1, const float* __restrict__ W2,
        const float* __restrict__ b2, const float* __restrict__ scale,
        const float* __restrict__ shift, float* __restrict__ out) {
    __shared__ v2f   sP[(D_EMB / 2) * D_EMB];  // 64 KB, paired rows
    __shared__ float sb[D_EMB];
    __shared__ float ssc[D_EMB];
    __shared__ float ssh[D_EMB];

    int tid = threadIdx.x;
    for (int i = tid; i < D_EMB * D_EMB; i += 256) {
        int k = i >> 7, n = i & 127, p = k >> 1;
        int nsw = n ^ ((p & 1) << 4);
        ((float*)&sP[p * D_EMB + nsw])[k & 1] = W2[i];
    }
    if (tid < D_EMB) {
        sb[tid]  = b2[tid];
        ssc[tid] = scale[tid];
        ssh[tid] = shift[tid];
    }
    __syncthreads();

    int wave = tid >> 5;
    int lane = tid & 31;
    int half = lane >> 4;
    int l16  = lane & 15;
    int row0 = blockIdx.x * 128 + wave * 16;
    int arow = row0 + l16;
    float amask = (arow < N_NODES) ? 1.0f : 0.0f;
    int    crow = (arow < N_NODES) ? arow : (N_NODES - 1);
    const v2f* h2 = (const v2f*)(h1 + (size_t)crow * D_EMB);

    v8f acc[8] = {};

    #pragma unroll 2
    for (int kk = 0; kk < D_EMB / 4; ++kk) {
        int p  = 2 * kk + half;
        int sw = p & 1;
        int c  = p * 2;
        v2f hv  = h2[p];
        v2f scv = *(const v2f*)&ssc[c];        // c even -> 8B aligned
        v2f shv = *(const v2f*)&ssh[c];
        v2f a;
        a.x = fmaxf(fmaf(hv.x, scv.x, shv.x), 0.0f) * amask;
        a.y = fmaxf(fmaf(hv.y, scv.y, shv.y), 0.0f) * amask;
        v2f bf[8];
        #pragma unroll
        for (int nt = 0; nt < 8; ++nt)
            bf[nt] = sP[p * D_EMB + ((nt ^ sw) * 16) + l16];
        #pragma unroll
        for (int nt = 0; nt < 8; ++nt)
            acc[nt] = __builtin_amdgcn_wmma_f32_16x16x4_f32(
                false, a, false, bf[nt], (short)0, acc[nt], false, false);
    }

    #pragma unroll
    for (int nt = 0; nt < 8; ++nt) {
        int col = nt * 16 + l16;
        float bias = sb[col];
        #pragma unroll
        for (int v = 0; v < 8; ++v) {
            int r = row0 + v + half * 8;
            if (r < N_NODES) {
                out[(size_t)r * D_EMB + col] = acc[nt][v] + bias;
            }
        }
    }
}

// ---------------------------------------------------------------------------
// Launcher
// ---------------------------------------------------------------------------
extern "C" void kernel_launch(void* const* d_in, const int* in_sizes, int n_in,
                              void* d_out, int out_size, void* d_ws, size_t ws_size,
                              hipStream_t stream) {
    (void)in_sizes; (void)n_in; (void)out_size; (void)ws_size;

    const float* x          = (const float*)d_in[0];
    const int*   edge_index = (const int*)d_in[1];
    // d_in[2] = edge_attr (unused by reference forward)
    const float* W1    = (const float*)d_in[3];
    const float* b1    = (const float*)d_in[4];
    const float* gamma = (const float*)d_in[5];
    const float* beta  = (const float*)d_in[6];
    const float* W2    = (const float*)d_in[7];
    const float* b2    = (const float*)d_in[8];
    float* out = (float*)d_out;

    // Workspace layout
    float* agg      = (float*)d_ws;                        // N*64
    float* h1       = agg + (size_t)N_NODES * D_IN;        // N*128
    float* colsum   = h1 + (size_t)N_NODES * D_EMB;        // 128
    float* colsumsq = colsum + D_EMB;                      // 128
    float* scale    = colsumsq + D_EMB;                    // 128
    float* shift    = scale + D_EMB;                       // 128

    // K0: zero agg + stats
    zero_ws_kernel<<<2048, 256, 0, stream>>>(agg, (long long)N_NODES * D_IN);
    zero_ws_kernel<<<1, 256, 0, stream>>>(colsum, 2 * D_EMB);

    // K1: scatter-add aggregation (float4 per thread)
    {
        long long work = (long long)N_EDGES * (D_IN / 4);
        int blocks = (int)((work + 255) / 256);
        scatter_add_kernel<<<blocks, 256, 0, stream>>>(x, edge_index, agg);
    }

    // K2: GEMM1 + fused BN stats
    int gblocks = (N_NODES + 127) / 128;
    gemm1_bn_kernel<<<gblocks, 256, 0, stream>>>(x, agg, W1, b1, h1,
                                                 colsum, colsumsq);

    // K3: finalize BN into scale/shift
    bn_finalize_kernel<<<1, D_EMB, 0, stream>>>(colsum, colsumsq, gamma, beta,
                                                scale, shift);

    // K4: BN-apply + ReLU + GEMM2
    gemm2_kernel<<<gblocks, 256, 0, stream>>>(h1, W2, b2, scale, shift, out);
}